// LossVariance_58334245814722
// MI455X (gfx1250) — compile-verified
//
#include <hip/hip_runtime.h>
#include <hip/hip_bf16.h>

// LossVariance on MI455X (gfx1250):
// memory-bound streaming reduction (~402 MB @ 23.3 TB/s => ~17 us floor).
// - b128 non-temporal loads from 12 coalesced channel streams
// - per-thread dual accumulators -> one v_wmma_f32_16x16x4_f32 (B = ones)
//   reduces all 64 wave partials in a single matrix op
// - deterministic two-pass reduction (per-block partials + finalize kernel)

typedef __attribute__((ext_vector_type(2))) float v2f;
typedef __attribute__((ext_vector_type(4))) float v4f;
typedef __attribute__((ext_vector_type(8))) float v8f;

#define LV_HW      (1u << 20)   // 1024*1024 pixels per (batch, channel)
#define LV_C       6
#define LV_B       8
#define LV_NBLK    256          // blocks per batch (grid.x)
#define LV_NTHR    256          // threads per block (8 wave32)

__device__ __forceinline__ float lv_wave_sum(float acc0, float acc1) {
#if defined(__gfx1250__) && __has_builtin(__builtin_amdgcn_wmma_f32_16x16x4_f32)
    // A (16x4 f32): lane m in [0,15] supplies K=0,1; lane m+16 supplies K=2,3.
    // With B = ones(4x16): D[m][n] = acc0[m]+acc1[m]+acc0[m+16]+acc1[m+16].
    // D layout: VGPR j holds rowsum[j] in lanes 0-15 and rowsum[j+8] in 16-31.
    v2f a; a.x = acc0; a.y = acc1;
    v2f bo; bo.x = 1.0f; bo.y = 1.0f;
    v8f c = {};
    v8f d = __builtin_amdgcn_wmma_f32_16x16x4_f32(
        /*neg_a=*/false, a, /*neg_b=*/false, bo,
        /*c_mod=*/(short)0, c, /*reuse_a=*/false, /*reuse_b=*/false);
    float h = d[0] + d[1] + d[2] + d[3] + d[4] + d[5] + d[6] + d[7];
    // lanes 0-15 hold sum(row 0..7), lanes 16-31 hold sum(row 8..15)
    h += __shfl_xor(h, 16, 32);
    return h;  // full 64-value wave sum, uniform across the wave
#else
    float v = acc0 + acc1;
    #pragma unroll
    for (int m = 16; m > 0; m >>= 1) v += __shfl_xor(v, m, 32);
    return v;
#endif
}

__global__ __launch_bounds__(LV_NTHR)
void lv_partial_kernel(const float* __restrict__ input,
                       const float* __restrict__ target,
                       float* __restrict__ part_sum,
                       unsigned* __restrict__ part_mask) {
    const unsigned b   = blockIdx.y;
    const unsigned tid = threadIdx.x;
    const size_t   base = (size_t)b * LV_C * (size_t)LV_HW;
    const float* __restrict__ in = input  + base;
    const float* __restrict__ tg = target + base;

    float    acc0 = 0.0f, acc1 = 0.0f;
    unsigned mask = 0u;

    const unsigned n4     = LV_HW / 4;            // 262144 float4 groups
    const unsigned stride = LV_NBLK * LV_NTHR;    // 65536 -> exactly 4 iters/thread

    for (unsigned i = blockIdx.x * LV_NTHR + tid; i < n4; i += stride) {
        const size_t p = (size_t)i * 4;
        v4f xv[LV_C], tv[LV_C];
        #pragma unroll
        for (int ch = 0; ch < LV_C; ++ch) {
            xv[ch] = __builtin_nontemporal_load(
                (const v4f*)(in + (size_t)ch * LV_HW + p));
            tv[ch] = __builtin_nontemporal_load(
                (const v4f*)(tg + (size_t)ch * LV_HW + p));
        }
        #pragma unroll
        for (int j = 0; j < 4; ++j) {
            float xs[LV_C], ts[LV_C];
            #pragma unroll
            for (int ch = 0; ch < LV_C; ++ch) {
                xs[ch] = xv[ch][j];
                ts[ch] = tv[ch][j];
            }
            // argmax over target channels; strict '>' keeps first max (jnp tie rule)
            int   label = 0;
            float mx    = ts[0];
            #pragma unroll
            for (int ch = 1; ch < LV_C; ++ch) {
                if (ts[ch] > mx) { mx = ts[ch]; label = ch; }
            }
            // unbiased variance over channels: (sum(x^2) - sum(x)^2/C) / (C-1)
            float s = 0.0f, sq = 0.0f;
            #pragma unroll
            for (int ch = 0; ch < LV_C; ++ch) {
                s  += xs[ch];
                sq  = fmaf(xs[ch], xs[ch], sq);
            }
            float var = (sq - s * s * (1.0f / 6.0f)) * (1.0f / 5.0f);
            if (label != 0) {
                mask |= 1u << label;
                if (j < 2) acc0 += var; else acc1 += var;
            }
        }
    }

    // wave-level OR of presence mask (order-independent -> deterministic)
    unsigned wm = mask;
    #pragma unroll
    for (int m = 16; m > 0; m >>= 1) wm |= __shfl_xor(wm, m, 32);

    // wave-level sum via WMMA (EXEC is all-1s here: uniform trip counts)
    float ws = lv_wave_sum(acc0, acc1);

    __shared__ float    sw[LV_NTHR / 32];
    __shared__ unsigned sm[LV_NTHR / 32];
    const unsigned wid = tid >> 5, lane = tid & 31;
    if (lane == 0) { sw[wid] = ws; sm[wid] = wm; }
    __syncthreads();

    if (tid == 0) {
        float    bs = 0.0f;
        unsigned bm = 0u;
        #pragma unroll
        for (int k = 0; k < LV_NTHR / 32; ++k) { bs += sw[k]; bm |= sm[k]; }
        part_sum[b * LV_NBLK + blockIdx.x]  = bs;
        part_mask[b * LV_NBLK + blockIdx.x] = bm;
    }
}

__global__ __launch_bounds__(LV_NBLK)
void lv_finalize_kernel(const float* __restrict__ part_sum,
                        const unsigned* __restrict__ part_mask,
                        float* __restrict__ out) {
    __shared__ float    sv[LV_NBLK];
    __shared__ unsigned sm[LV_NBLK];
    const unsigned tid = threadIdx.x;
    float loss_acc = 0.0f;

    for (int b = 0; b < LV_B; ++b) {
        sv[tid] = part_sum[b * LV_NBLK + tid];
        sm[tid] = part_mask[b * LV_NBLK + tid];
        __syncthreads();
        // fixed binary-tree reduction -> deterministic fp summation order
        for (unsigned off = LV_NBLK / 2; off > 0; off >>= 1) {
            if (tid < off) { sv[tid] += sv[tid + off]; sm[tid] |= sm[tid + off]; }
            __syncthreads();
        }
        if (tid == 0) {
            float n_unique = (float)__popc(sm[0] & 0x3Eu);  // labels 1..5
            loss_acc += sv[0] / (n_unique + 1e-8f);
        }
        __syncthreads();
    }
    if (tid == 0) out[0] = loss_acc * (1.0f / (float)LV_B);
}

extern "C" void kernel_launch(void* const* d_in, const int* in_sizes, int n_in,
                              void* d_out, int out_size, void* d_ws, size_t ws_size,
                              hipStream_t stream) {
    (void)in_sizes; (void)n_in; (void)out_size; (void)ws_size;
    const float* input  = (const float*)d_in[0];
    const float* target = (const float*)d_in[1];
    float*       out    = (float*)d_out;

    // workspace: [B*NBLK] float partial sums, then [B*NBLK] u32 partial masks
    float*    part_sum  = (float*)d_ws;
    unsigned* part_mask = (unsigned*)((char*)d_ws + (size_t)LV_B * LV_NBLK * sizeof(float));

    dim3 grid(LV_NBLK, LV_B, 1);
    dim3 block(LV_NTHR, 1, 1);
    lv_partial_kernel<<<grid, block, 0, stream>>>(input, target, part_sum, part_mask);
    lv_finalize_kernel<<<1, dim3(LV_NBLK, 1, 1), 0, stream>>>(part_sum, part_mask, out);
}